// SpectralEmbedding_58548994179739
// MI455X (gfx1250) — compile-verified
//
#include <hip/hip_runtime.h>
#include <hip/hip_bf16.h>
#include <math.h>

#define TILE   64
#define OVR    16
#define STRIDEP 48
#define NTS    7
#define NT     49
#define HH     352
#define WW     352
#define NPIX   4096
#define MITER  30
#define NEVEC  4

typedef __attribute__((ext_vector_type(2)))  float     v2f;
typedef __attribute__((ext_vector_type(4)))  float     v4f;
typedef __attribute__((ext_vector_type(8)))  float     v8f;
typedef __attribute__((ext_vector_type(4)))  _Float16  v4h;
typedef __attribute__((ext_vector_type(4)))  unsigned  u32x4;
typedef __attribute__((ext_vector_type(4)))  int       i32x4;
typedef __attribute__((ext_vector_type(8)))  int       i32x8;

// ---- unique-offset tap table (collapses the reference's per-radius duplicates) ----
struct Taps { int dy[128]; int dx[128]; float w[128]; int n; };
constexpr Taps make_taps() {
    Taps t{};
    t.n = 0;
    const int   rr[6] = {1, 2, 3, 4, 5, 6};
    const float rw[6] = {1.0f, 0.6f, 0.4f, 0.3f, 0.2f, 0.1f};
    for (int dy = -6; dy <= 6; ++dy)
        for (int dx = -6; dx <= 6; ++dx) {
            if (dy == 0 && dx == 0) continue;
            int d2 = dy * dy + dx * dx;
            if (d2 > 36) continue;
            float w = 0.f;
            for (int r = 0; r < 6; ++r)
                if (d2 <= rr[r] * rr[r]) w += rw[r];
            t.dy[t.n] = dy; t.dx[t.n] = dx; t.w[t.n] = w; ++t.n;
        }
    return t;
}
static constexpr Taps c_taps = make_taps();
constexpr int KU = make_taps().n;           // 112 unique taps
static_assert(KU <= 128, "tap table overflow");

__device__ __forceinline__ float hash_unit(unsigned x) {
    x = x * 747796405u + 2891336453u;
    x = ((x >> ((x >> 28) + 4u)) ^ x) * 277803737u;
    x = (x >> 22) ^ x;
    return (float)x * (1.0f / 4294967296.0f) * 2.0f - 1.0f;   // (-1, 1)
}

// =====================================================================
// Kernel A: TDM (tensor_load_to_lds) pulls the strided 64x(64*3f) tile
// into LDS; then per-tile edge weights (fp16, L2-resident) + diagonal.
// =====================================================================
__global__ void edge_kernel(const float* __restrict__ img,
                            _Float16* __restrict__ ew,
                            float* __restrict__ deg) {
    __shared__ float col[TILE * TILE * 3];        // AoS: col[p*3 + c]
    const int tile = blockIdx.x;
    const int tid  = threadIdx.x;                 // 256 threads
    const int oy = (tile / NTS) * STRIDEP;
    const int ox = (tile % NTS) * STRIDEP;

    if (tid == 0) {
        // ---- Tensor DMA descriptor: 2D tile, 192 x 64 elements of 4B,
        //      row stride 1056 elements, into contiguous LDS. ----
        unsigned ldsbase = (unsigned)(uintptr_t)(void*)col;   // flat low 32b = LDS offset
        const float* gsrc = img + ((size_t)oy * WW + ox) * 3;
        unsigned long long ga = (unsigned long long)(uintptr_t)gsrc;
        u32x4 g0 = { 1u,                                   // count=1 (valid descriptor)
                     ldsbase,                              // lds_addr
                     (unsigned)(ga & 0xffffffffull),       // global_addr[31:0]
                     (unsigned)((ga >> 32) & 0x01ffffffull) | (2u << 30) }; // [56:32] | type=2
        i32x8 g1 = { (int)(2u << 16),          // data_size=2 (4B), mask=0, no flags
                     (int)(1056u << 16),       // tensor_dim0[15:0] in [31:16]
                     (int)(64u << 16),         // tensor_dim0[31:16]=0 | tensor_dim1[15:0]
                     (int)(192u << 16),        // tensor_dim1[31:16]=0 | tile_dim0=192
                     64,                       // tile_dim1=64 | tile_dim2=0
                     1056,                     // tensor_dim0_stride[31:0]
                     0, 0 };                   // stride hi, tensor_dim1_stride
        i32x4 gz4 = { 0, 0, 0, 0 };
#if defined(__clang_major__) && (__clang_major__ >= 23)
        i32x8 gz8 = { 0, 0, 0, 0, 0, 0, 0, 0 };
        __builtin_amdgcn_tensor_load_to_lds(g0, g1, gz4, gz4, gz8, 0);
#else
        __builtin_amdgcn_tensor_load_to_lds(g0, g1, gz4, gz4, 0);
#endif
        __builtin_amdgcn_s_wait_tensorcnt(0);
    }
    __syncthreads();

    _Float16* ewt = ew + (size_t)tile * KU * NPIX;
    for (int i = 0; i < 4; ++i) {
        int p0  = (tid + i * 256) * 4;            // 4 consecutive pixels (same row)
        int py  = p0 >> 6, px0 = p0 & 63;
        float cc[4][3];
        #pragma unroll
        for (int j = 0; j < 4; ++j) {
            cc[j][0] = col[(p0 + j) * 3 + 0];
            cc[j][1] = col[(p0 + j) * 3 + 1];
            cc[j][2] = col[(p0 + j) * 3 + 2];
        }
        float dsum[4] = {0.f, 0.f, 0.f, 0.f};
        for (int k = 0; k < KU; ++k) {
            int ny = py + c_taps.dy[k];
            int dxk = c_taps.dx[k];
            float wk = c_taps.w[k];
            v4h pack;
            bool rowok = (ny >= 0) & (ny < TILE);
            #pragma unroll
            for (int j = 0; j < 4; ++j) {
                int nx = px0 + j + dxk;
                float v = 0.f;
                if (rowok && nx >= 0 && nx < TILE) {
                    int q = ny * TILE + nx;
                    float dr = col[q * 3 + 0] - cc[j][0];
                    float dg = col[q * 3 + 1] - cc[j][1];
                    float db = col[q * 3 + 2] - cc[j][2];
                    float d  = sqrtf(dr * dr + dg * dg + db * db);
                    v = wk * __expf(d * (-1.0f / 0.15f));
                }
                pack[j] = (_Float16)v;
                dsum[j] += v;
            }
            *(v4h*)(ewt + (size_t)k * NPIX + p0) = pack;   // 8B/lane, contiguous
        }
        v4f dv = { dsum[0], dsum[1], dsum[2], dsum[3] };
        *(v4f*)(deg + (size_t)tile * NPIX + p0) = dv;
    }
}

// =====================================================================
// Kernel B: Lanczos (30 iters) — one workgroup per tile, x kept in LDS.
// ew streamed as v4h (b64/lane, fully coalesced) from the L2-resident slab.
// =====================================================================
__device__ __forceinline__ float block_reduce(float v, float* red, int tid) {
    for (int o = 16; o > 0; o >>= 1) v += __shfl_xor(v, o, 32);
    if (tid == 0) *red = 0.f;
    __syncthreads();
    if ((tid & 31) == 0) atomicAdd(red, v);
    __syncthreads();
    float r = *red;
    __syncthreads();
    return r;
}

__global__ void lanczos_kernel(const _Float16* __restrict__ ew,
                               const float* __restrict__ deg,
                               float* __restrict__ Q,
                               float* __restrict__ alb) {
    __shared__ float xlds[NPIX];
    __shared__ float red;
    const int tile = blockIdx.x;
    const int tid  = threadIdx.x;                 // 1024 threads
    const int p0   = tid * 4;                     // 4 consecutive pixels per thread
    const _Float16* ewt  = ew  + (size_t)tile * KU * NPIX;
    float* Qt   = Q   + (size_t)tile * MITER * NPIX;
    float* albt = alb + (size_t)tile * 64;

    v4f dg = *(const v4f*)(deg + (size_t)tile * NPIX + p0);
    float xi[4], vp[4];
    float s0 = 0.f;
    #pragma unroll
    for (int i = 0; i < 4; ++i) {
        float r = hash_unit((unsigned)(tile * NPIX + p0 + i) * 2654435761u + 7u);
        xi[i] = r; vp[i] = 0.f;
        s0 += r;
    }
    // deflate against constant vector, then normalize
    float sum0 = block_reduce(s0, &red, tid);
    float s2 = 0.f;
    #pragma unroll
    for (int i = 0; i < 4; ++i) { xi[i] -= sum0 * (1.0f / NPIX); s2 += xi[i] * xi[i]; }
    float nrm2 = block_reduce(s2, &red, tid);
    float inv  = 1.0f / sqrtf(fmaxf(nrm2, 1e-20f));
    {
        v4f xv;
        #pragma unroll
        for (int i = 0; i < 4; ++i) { xi[i] *= inv; xv[i] = xi[i]; }
        *(v4f*)(xlds + p0) = xv;
    }
    __syncthreads();

    float beta = 0.f;
    for (int m = 0; m < MITER; ++m) {
        float wv[4];
        #pragma unroll
        for (int i = 0; i < 4; ++i) wv[i] = dg[i] * xi[i];
        for (int k = 0; k < KU; ++k) {
            if ((k & 7) == 0 && k + 8 < KU)
                __builtin_prefetch(ewt + (size_t)(k + 8) * NPIX + p0, 0, 1);
            int off = c_taps.dy[k] * TILE + c_taps.dx[k];
            v4h e = *(const v4h*)(ewt + (size_t)k * NPIX + p0);
            #pragma unroll
            for (int i = 0; i < 4; ++i) {
                int q = p0 + i + off;
                q = q < 0 ? 0 : (q > NPIX - 1 ? NPIX - 1 : q);   // ew==0 where invalid
                wv[i] -= (float)e[i] * xlds[q];
            }
        }
        #pragma unroll
        for (int i = 0; i < 4; ++i) wv[i] -= beta * vp[i];

        float pa = 0.f;
        #pragma unroll
        for (int i = 0; i < 4; ++i) pa += wv[i] * xi[i];
        float alpha = block_reduce(pa, &red, tid);
        float ps = 0.f;
        #pragma unroll
        for (int i = 0; i < 4; ++i) { wv[i] -= alpha * xi[i]; ps += wv[i]; }
        float swv = block_reduce(ps, &red, tid);     // deflation: w -= c*(c@w)
        float pn = 0.f;
        #pragma unroll
        for (int i = 0; i < 4; ++i) { wv[i] -= swv * (1.0f / NPIX); pn += wv[i] * wv[i]; }
        float b2 = block_reduce(pn, &red, tid);
        float b  = sqrtf(fmaxf(b2, 0.f));
        float binv = 1.0f / fmaxf(b, 1e-10f);

        {
            v4f qv = { xi[0], xi[1], xi[2], xi[3] };
            *(v4f*)(Qt + (size_t)m * NPIX + p0) = qv;       // b128 store
        }
        if (tid == 0) { albt[m] = alpha; albt[32 + m] = b; }

        __syncthreads();
        {
            v4f xv;
            #pragma unroll
            for (int i = 0; i < 4; ++i) {
                vp[i] = xi[i];
                xi[i] = wv[i] * binv;
                xv[i] = xi[i];
            }
            *(v4f*)(xlds + p0) = xv;                        // b128 LDS store
        }
        __syncthreads();
        beta = b;
    }
}

// =====================================================================
// Kernel C: 30x30 tridiagonal eigh via cyclic Jacobi (serial per tile)
// writes zero-padded U (32 x 16, only [0:30, 0:4] nonzero) + zeroed sums
// =====================================================================
__global__ void eig_kernel(const float* __restrict__ alb,
                           float* __restrict__ U4p,
                           float* __restrict__ sums) {
    __shared__ float A[MITER][MITER];
    __shared__ float V[MITER][MITER];
    const int tile = blockIdx.x;
    if (threadIdx.x != 0) return;
    const float* a = alb + (size_t)tile * 64;

    for (int i = 0; i < MITER; ++i)
        for (int j = 0; j < MITER; ++j) { A[i][j] = 0.f; V[i][j] = (i == j) ? 1.f : 0.f; }
    for (int i = 0; i < MITER; ++i) A[i][i] = a[i];
    for (int i = 0; i < MITER - 1; ++i) { A[i][i + 1] = a[32 + i]; A[i + 1][i] = a[32 + i]; }

    for (int sweep = 0; sweep < 24; ++sweep) {
        for (int p = 0; p < MITER - 1; ++p)
            for (int q = p + 1; q < MITER; ++q) {
                float apq = A[p][q];
                if (fabsf(apq) < 1e-12f) continue;
                float theta = (A[q][q] - A[p][p]) / (2.0f * apq);
                float t = 1.0f / (fabsf(theta) + sqrtf(theta * theta + 1.0f));
                if (theta < 0.f) t = -t;
                float c = 1.0f / sqrtf(t * t + 1.0f);
                float s = t * c;
                for (int k = 0; k < MITER; ++k) {
                    float akp = A[k][p], akq = A[k][q];
                    A[k][p] = c * akp - s * akq;
                    A[k][q] = s * akp + c * akq;
                }
                for (int k = 0; k < MITER; ++k) {
                    float apk = A[p][k], aqk = A[q][k];
                    A[p][k] = c * apk - s * aqk;
                    A[q][k] = s * apk + c * aqk;
                }
                for (int k = 0; k < MITER; ++k) {
                    float vkp = V[k][p], vkq = V[k][q];
                    V[k][p] = c * vkp - s * vkq;
                    V[k][q] = s * vkp + c * vkq;
                }
            }
    }
    // select 4 smallest eigenvalues (ascending, like eigh)
    bool used[MITER] = {};
    int  idx[NEVEC];
    for (int j = 0; j < NEVEC; ++j) {
        int best = -1; float bv = 1e30f;
        for (int i = 0; i < MITER; ++i)
            if (!used[i] && A[i][i] < bv) { bv = A[i][i]; best = i; }
        used[best] = true; idx[j] = best;
    }
    float* U = U4p + (size_t)tile * 32 * 16;
    for (int k = 0; k < 32; ++k)
        for (int n = 0; n < 16; ++n) U[k * 16 + n] = 0.f;
    for (int k = 0; k < MITER; ++k)
        for (int j = 0; j < NEVEC; ++j) U[k * 16 + j] = V[k][idx[j]];
    for (int e = 0; e < NEVEC; ++e) sums[tile * NEVEC + e] = 0.f;
}

// =====================================================================
// Kernel D: ritz = Q^T (4096x30) @ U (30x4)  via V_WMMA_F32_16X16X4_F32
//   K padded 30 -> 32 (8 k-chunks of 4); B zero-padded to 16 columns.
//   A 16x4 layout:  lanes0-15 M=0..15 {K=0,1}; lanes16-31 M=0..15 {K=2,3}
//   B 4x16 layout:  v0: K=0|K=2 across half-waves; v1: K=1|K=3
//   C/D 16x16:      VGPR j -> M=j (+8 for upper half-wave), N=lane&15
// =====================================================================
__global__ void ritz_kernel(const float* __restrict__ Q,
                            const float* __restrict__ U4p,
                            float* __restrict__ ritz,
                            float* __restrict__ sums) {
    const int tile = blockIdx.x;
    const int lane = threadIdx.x & 31;
    const int wave = threadIdx.x >> 5;            // 8 waves of 32
    const int n    = lane & 15;
    const int hi   = lane >> 4;                   // half-wave select

    const float* Qt = Q   + (size_t)tile * MITER * NPIX;
    const float* U  = U4p + (size_t)tile * 32 * 16;
    float*       rt = ritz + (size_t)tile * NPIX * NEVEC;

    v2f breg[8];
    #pragma unroll
    for (int c8 = 0; c8 < 8; ++c8) {
        int k = c8 * 4 + hi * 2;
        v2f b; b.x = U[k * 16 + n]; b.y = U[(k + 1) * 16 + n];
        breg[c8] = b;
    }

    float colsum = 0.f;
    for (int pb = wave; pb < NPIX / 16; pb += 8) {
        int p = pb * 16 + n;                      // pixel row M for this lane
        v8f acc = {};
        #pragma unroll
        for (int c8 = 0; c8 < 8; ++c8) {
            int k0 = c8 * 4 + hi * 2;
            int r0 = k0     < MITER ? k0     : MITER - 1;   // padded rows hit B==0
            int r1 = k0 + 1 < MITER ? k0 + 1 : MITER - 1;
            v2f a;
            a.x = Qt[(size_t)r0 * NPIX + p];
            a.y = Qt[(size_t)r1 * NPIX + p];
            acc = __builtin_amdgcn_wmma_f32_16x16x4_f32(
                false, a, false, breg[c8], (short)0, acc, false, false);
        }
        if (n < NEVEC) {
            #pragma unroll
            for (int j = 0; j < 8; ++j) {
                int m = j + hi * 8;
                float v = acc[j];
                rt[(size_t)(pb * 16 + m) * NEVEC + n] = v;
                colsum += v;
            }
        }
    }
    if (n < NEVEC) atomicAdd(&sums[tile * NEVEC + n], colsum);
}

// =====================================================================
// Kernel E: sign-fix + linear-taper blend into (H, W, 4)
// =====================================================================
__global__ void blend_kernel(const float* __restrict__ ritz,
                             const float* __restrict__ sums,
                             float* __restrict__ out) {
    int idx = blockIdx.x * blockDim.x + threadIdx.x;
    if (idx >= HH * WW) return;
    int y = idx / WW, x = idx % WW;
    float num[NEVEC] = {0.f, 0.f, 0.f, 0.f};
    float den = 0.f;
    for (int ty = 0; ty < NTS; ++ty) {
        int ly = y - ty * STRIDEP;
        if (ly < 0 || ly >= TILE) continue;
        float t1y = fminf(fminf((ly + 1) * (1.0f / OVR), (TILE - ly) * (1.0f / OVR)), 1.0f);
        for (int tx = 0; tx < NTS; ++tx) {
            int lx = x - tx * STRIDEP;
            if (lx < 0 || lx >= TILE) continue;
            float w2 = t1y * fminf(fminf((lx + 1) * (1.0f / OVR), (TILE - lx) * (1.0f / OVR)), 1.0f);
            int tile = ty * NTS + tx;
            const float* r = ritz + ((size_t)tile * NPIX + ly * TILE + lx) * NEVEC;
            #pragma unroll
            for (int e = 0; e < NEVEC; ++e) {
                float s = sums[tile * NEVEC + e] >= 0.f ? 1.0f : -1.0f;
                num[e] += w2 * r[e] * s;
            }
            den += w2;
        }
    }
    float inv = 1.0f / den;
    #pragma unroll
    for (int e = 0; e < NEVEC; ++e) out[(size_t)idx * NEVEC + e] = num[e] * inv;
}

// =====================================================================
extern "C" void kernel_launch(void* const* d_in, const int* in_sizes, int n_in,
                              void* d_out, int out_size, void* d_ws, size_t ws_size,
                              hipStream_t stream) {
    const float* image = (const float*)d_in[0];
    float* out = (float*)d_out;

    char* ws = (char*)d_ws;
    auto align256 = [](size_t v) { return (v + 255) & ~(size_t)255; };
    size_t off = 0;
    _Float16* ew = (_Float16*)(ws + off); off = align256(off + (size_t)NT * KU * NPIX * sizeof(_Float16));
    float* deg  = (float*)(ws + off);     off = align256(off + (size_t)NT * NPIX * sizeof(float));
    float* Q    = (float*)(ws + off);     off = align256(off + (size_t)NT * MITER * NPIX * sizeof(float));
    float* alb  = (float*)(ws + off);     off = align256(off + (size_t)NT * 64 * sizeof(float));
    float* U4p  = (float*)(ws + off);     off = align256(off + (size_t)NT * 32 * 16 * sizeof(float));
    float* sums = (float*)(ws + off);     off = align256(off + (size_t)NT * NEVEC * sizeof(float));
    float* ritz = (float*)(ws + off);     off = align256(off + (size_t)NT * NPIX * NEVEC * sizeof(float));
    (void)ws_size; (void)in_sizes; (void)n_in; (void)out_size;   // ~73 MB total scratch

    edge_kernel   <<<NT, 256,  0, stream>>>(image, ew, deg);
    lanczos_kernel<<<NT, 1024, 0, stream>>>(ew, deg, Q, alb);
    eig_kernel    <<<NT, 32,   0, stream>>>(alb, U4p, sums);
    ritz_kernel   <<<NT, 256,  0, stream>>>(Q, U4p, ritz, sums);
    blend_kernel  <<<(HH * WW + 255) / 256, 256, 0, stream>>>(ritz, sums, out);
}